// CRF_7773890806365
// MI455X (gfx1250) — compile-verified
//
#include <hip/hip_runtime.h>

// CRF forward (logsumexp scan) on gfx1250 using V_WMMA_F32_16X16X4_F32.
//
// Exp-domain scan, exact power-of-two rescaling every 4 steps:
//   invariant: alpha[b,i] = Eacc_b*ln2 + log(y[b,i])
//   step:      S = (P_rho . E) . y   (E = exp(transitions); WMMA, 2 tiles x 8 K-slices)
//              y' = S * exp(feat_t);  periodically y *= 2^-e, Eacc += e
//
// The row permutation rho makes the WMMA D-register layout IDENTICAL to the
// B-fragment layout needed by the next step, so the scan loop has zero
// cross-lane data movement (no ds_bpermute in the steady state).

typedef __attribute__((ext_vector_type(2))) float v2f;
typedef __attribute__((ext_vector_type(8))) float v8f;

#define CRF_L 32
#define CRF_LN2 0.6931471805599453f

// rho: D-position row (M = 16R + v' + 8h) -> logical CRF state held there.
__device__ __forceinline__ int crf_rho(int M) {
  const int R = M >> 4, r = M & 15, hh = r >> 3, vv = r & 7;
  return (R << 4) + 4 * (vv >> 1) + (vv & 1) + 2 * hh;
}

__global__ __launch_bounds__(32)
void crf_forward_kernel(const float* __restrict__ features,
                        const float* __restrict__ transitions,
                        float* __restrict__ fwd, int Tlen) {
  const int lane = threadIdx.x & 31;
  const int n = lane & 15;      // batch column (matrix N index / A-fragment M index)
  const int h = lane >> 4;      // wave half
  const int b = blockIdx.x * 16 + n;

  // ---- A fragments: E rows pre-permuted by rho ---------------------------
  // A 16x4 f32 layout: lanes 0-15 hold K={0,1}, lanes 16-31 K={2,3}; M = lane&15.
  v2f eA[2][8];
  {
    const int rowg[2] = { crf_rho(n), crf_rho(16 + n) };
#pragma unroll
    for (int R = 0; R < 2; ++R) {
#pragma unroll
      for (int s = 0; s < 8; ++s) {
        const int col = 4 * s + 2 * h;
        eA[R][s][0] = __expf(transitions[rowg[R] * CRF_L + col + 0]);
        eA[R][s][1] = __expf(transitions[rowg[R] * CRF_L + col + 1]);
      }
    }
  }

  // ---- state y, stored directly as the 8 B-fragments (v2f each) ----------
  // Position (u = 4R+j, v) in half h holds state 16R + 4j + v + 2h.
  v2f yp[8];
#pragma unroll
  for (int u = 0; u < 8; ++u) { yp[u][0] = 0.0f; yp[u][1] = 0.0f; }
  // alpha0: 0 at START state 0 (position u=0,v=0,h=0), ~-inf elsewhere.
  yp[0][0] = (h == 0) ? 1.0f : 0.0f;
  int Eacc = 0;

  // ---- feature stream: 8 x b64 per step, states 16R+4j+2h+{0,1} ----------
  const v2f* frow = (const v2f*)(features + (size_t)b * Tlen * CRF_L);
  v2f fc[8];
#pragma unroll
  for (int u = 0; u < 8; ++u) fc[u] = frow[16 * 1 + 2 * u + h];

  for (int t = 1; t < Tlen; ++t) {
    // software-pipeline next timestep's feature loads
    v2f gc[8];
    if (t + 1 < Tlen) {
#pragma unroll
      for (int u = 0; u < 8; ++u) gc[u] = frow[16 * (t + 1) + 2 * u + h];
    }
    if (t + 8 < Tlen)
      __builtin_prefetch(features + ((size_t)b * Tlen + t + 8) * CRF_L + 2 * h, 0, 0);

    // w = exp(feat), same register positions as D/yp
    v2f wp[8];
#pragma unroll
    for (int u = 0; u < 8; ++u) {
      wp[u][0] = __expf(fc[u][0]);
      wp[u][1] = __expf(fc[u][1]);
    }

    // ---- S = E' . y : 16x WMMA, dual accumulators per tile ---------------
    v8f acc[2][2];
#pragma unroll
    for (int R = 0; R < 2; ++R)
#pragma unroll
      for (int p = 0; p < 2; ++p)
#pragma unroll
        for (int v = 0; v < 8; ++v) acc[R][p][v] = 0.0f;

#pragma unroll
    for (int s = 0; s < 8; ++s) {
      const int p = s & 1;
      acc[0][p] = __builtin_amdgcn_wmma_f32_16x16x4_f32(false, eA[0][s], false, yp[s],
                                                        (short)0, acc[0][p], false, false);
      acc[1][p] = __builtin_amdgcn_wmma_f32_16x16x4_f32(false, eA[1][s], false, yp[s],
                                                        (short)0, acc[1][p], false, false);
    }

    // ---- y' = S * w : D registers feed next step's B fragments directly --
#pragma unroll
    for (int u = 0; u < 8; ++u) {
      const int R = u >> 2;
      const int j = u & 3;
      yp[u][0] = (acc[R][0][2 * j + 0] + acc[R][1][2 * j + 0]) * wp[u][0];
      yp[u][1] = (acc[R][0][2 * j + 1] + acc[R][1][2 * j + 1]) * wp[u][1];
    }

    // ---- exact power-of-two renorm, amortized over 4 steps ---------------
    // Per-step growth <= ~2^19, so 4 unnormalized steps stay far below 2^127.
    if ((t & 3) == 0) {
      float m = 0.0f;
#pragma unroll
      for (int u = 0; u < 8; ++u) m = fmaxf(m, fmaxf(yp[u][0], yp[u][1]));
      m = fmaxf(m, __shfl_xor(m, 16));
      int e;
      (void)frexpf(m, &e);                  // m = mant * 2^e
      const float sc = ldexpf(1.0f, -e);    // exact
      Eacc += e;
#pragma unroll
      for (int u = 0; u < 8; ++u) { yp[u][0] *= sc; yp[u][1] *= sc; }
    }

#pragma unroll
    for (int u = 0; u < 8; ++u) fc[u] = gc[u];
  }

  // forward_score[b] = Eacc*ln2 + log(sum_i y_i)
  float ssum = 0.0f;
#pragma unroll
  for (int u = 0; u < 8; ++u) ssum += yp[u][0] + yp[u][1];
  ssum += __shfl_xor(ssum, 16);
  const float fscore = (float)Eacc * CRF_LN2 + logf(ssum);
  if (h == 0) fwd[b] = fscore;
}

__global__ __launch_bounds__(256)
void crf_gold_kernel(const float* __restrict__ features,
                     const float* __restrict__ transitions,
                     const int* __restrict__ labels,
                     float* __restrict__ gold, int Tlen) {
  const int b = blockIdx.x;
  const int* lb = labels + (size_t)b * Tlen;
  const float* fb = features + (size_t)b * Tlen * CRF_L;
  float acc = 0.0f;
  for (int t = 1 + threadIdx.x; t < Tlen; t += blockDim.x) {
    const int cur = lb[t];
    const int prev = lb[t - 1];
    acc += transitions[cur * CRF_L + prev] + fb[t * CRF_L + cur];
  }
  __shared__ float red[256];
  red[threadIdx.x] = acc;
  __syncthreads();
  for (int off = 128; off > 0; off >>= 1) {
    if ((int)threadIdx.x < off) red[threadIdx.x] += red[threadIdx.x + off];
    __syncthreads();
  }
  if (threadIdx.x == 0) gold[b] = red[0];
}

__global__ __launch_bounds__(256)
void crf_final_kernel(const float* __restrict__ fwd,
                      const float* __restrict__ gold,
                      float* __restrict__ out, int B) {
  float acc = 0.0f;
  for (int b = threadIdx.x; b < B; b += 256) acc += fwd[b] - gold[b];
  __shared__ float red[256];
  red[threadIdx.x] = acc;
  __syncthreads();
  for (int off = 128; off > 0; off >>= 1) {
    if ((int)threadIdx.x < off) red[threadIdx.x] += red[threadIdx.x + off];
    __syncthreads();
  }
  if (threadIdx.x == 0) out[0] = red[0] / (float)B;
}

extern "C" void kernel_launch(void* const* d_in, const int* in_sizes, int n_in,
                              void* d_out, int out_size, void* d_ws, size_t ws_size,
                              hipStream_t stream) {
  const float* features    = (const float*)d_in[0];
  const float* transitions = (const float*)d_in[1];
  const int*   labels      = (const int*)d_in[2];

  const int Tlen = 1024;
  const int B = in_sizes[2] / Tlen;   // label_ids is (B, T)

  float* fwd  = (float*)d_ws;
  float* gold = fwd + B;

  crf_forward_kernel<<<B / 16, 32, 0, stream>>>(features, transitions, fwd, Tlen);
  crf_gold_kernel<<<B, 256, 0, stream>>>(features, transitions, labels, gold, Tlen);
  crf_final_kernel<<<1, 256, 0, stream>>>(fwd, gold, (float*)d_out, B);
}